// CTCLossWithLabelPriors_87754771792505
// MI455X (gfx1250) — compile-verified
//
#include <hip/hip_runtime.h>

#define NEG_INF (-1e30f)

constexpr int T_DIM = 1000;
constexpr int N_DIM = 32;
constexpr int C_DIM = 1024;
constexpr int S_DIM = 200;
constexpr int L_DIM = 2 * S_DIM + 1;   // 401
constexpr int BLOCK = 512;             // 16 wave32s; 512 thr * 2 floats = 1024 = one row
constexpr int NC    = N_DIM * C_DIM;   // float stride between consecutive t rows

// ---- CDNA5 async global->LDS path (ASYNCcnt), guarded for toolchain drift ----
#if defined(__has_builtin)
#if __has_builtin(__builtin_amdgcn_global_load_async_to_lds_b64) && \
    __has_builtin(__builtin_amdgcn_s_wait_asynccnt)
#define CTC_ASYNC 1
#endif
#endif

#ifdef CTC_ASYNC
typedef int v2i __attribute__((ext_vector_type(2)));
typedef __attribute__((address_space(1))) v2i* gptr_v2i;  // global (prints as __device__)
typedef __attribute__((address_space(3))) v2i* lptr_v2i;  // LDS    (prints as __shared__)
#define ROW_COPY(gsrc, ldst)                                                  \
  __builtin_amdgcn_global_load_async_to_lds_b64(                              \
      (gptr_v2i)(void*)(gsrc), (lptr_v2i)(void*)(ldst), 0, 0)
#define WAIT_ASYNC(N) __builtin_amdgcn_s_wait_asynccnt(N)
#else
#define ROW_COPY(gsrc, ldst) (*(float2*)(ldst) = *(const float2*)(gsrc))
#define WAIT_ASYNC(N) ((void)0)
#endif

// logaddexp: m + log(1 + exp(min-max)); NEG_INF-safe (exp underflows to 0)
__device__ __forceinline__ float lae(float a, float b) {
  float m = fmaxf(a, b);
  float d = fminf(a, b) - m;
  return m + __logf(1.0f + __expf(d));
}

__global__ __launch_bounds__(BLOCK) void ctc_alpha_kernel(
    const float* __restrict__ log_probs,      // (T, N, C)
    const int*   __restrict__ targets,        // (N, S)
    const int*   __restrict__ input_lengths,  // (N,)
    const int*   __restrict__ target_lengths, // (N,)
    float*       __restrict__ ws)             // (N,) per-sample -log p
{
  __shared__ float row[3][C_DIM];        // 3-deep async prefetch ring (12 KB)
  __shared__ float aL[2][L_DIM + 3];     // ping-pong alpha, 2-elem NEG_INF front pad

  const int n   = blockIdx.x;
  const int tid = threadIdx.x;
  const int inLen = input_lengths[n];
  const int tl    = target_lengths[n];

  // Per-lane extended label index and skip permission (l odd -> label, even -> blank)
  int  myidx = 0;     // blank
  bool allow = false;
  if (tid < L_DIM && (tid & 1)) {
    const int s = tid >> 1;
    myidx = targets[n * S_DIM + s];
    const int prev = (s > 0) ? targets[n * S_DIM + s - 1] : 0;
    allow = (myidx != 0) && (myidx != prev);
  }

  // Per-thread 8-byte slice of each (t, n, :) row
  const float* grow = log_probs + (size_t)n * C_DIM + 2 * tid;

  // Prologue: launch rows 0,1,2; make row 0 visible
  ROW_COPY(grow + 0 * NC, &row[0][2 * tid]);
  ROW_COPY(grow + 1 * NC, &row[1][2 * tid]);
  ROW_COPY(grow + 2 * NC, &row[2][2 * tid]);
  WAIT_ASYNC(2);
  __syncthreads();

  // alpha0
  float a_self = NEG_INF;
  if (tid < L_DIM) {
    if (tid == 0)      a_self = row[0][0];                           // blank
    else if (tid == 1) a_self = (tl > 0) ? row[0][myidx] : NEG_INF;  // first label
    aL[0][2 + tid] = a_self;
  }
  if (tid < 2) { aL[0][tid] = NEG_INF; aL[1][tid] = NEG_INF; }       // front pads

  const float* gpre = grow + 3 * (size_t)NC;  // row t+2 pointer at t==1

  for (int t = 1; t < T_DIM; ++t) {
    // own slice of row t complete (in-order async completion), then all slices + alpha stores
    if (t < T_DIM - 1) { WAIT_ASYNC(1); } else { WAIT_ASYNC(0); }
    __syncthreads();

    const int c  = t % 3;
    const int pr = (t + 1) & 1;  // read parity
    const int pw = t & 1;        // write parity

    if (tid < L_DIM) {
      const float lp = row[c][myidx];                  // LDS gather
      const float a2 = aL[pr][tid + 1];                // alpha[l-1]
      const float a3 = allow ? aL[pr][tid] : NEG_INF;  // alpha[l-2]
      const float nv = lae(lae(a_self, a2), a3) + lp;
      a_self = (t < inLen) ? nv : a_self;              // freeze past input length
      aL[pw][tid + 2] = a_self;
    }

    if (t + 2 < T_DIM) {          // prefetch row t+2 into buffer last read at t-1
      ROW_COPY(gpre, &row[(t + 2) % 3][2 * tid]);
      gpre += NC;
    }
  }
  __syncthreads();

  if (tid == 0) {
    const int pf = (T_DIM - 1) & 1;
    const int l1 = 2 * tl;
    const int l2 = (l1 - 1 > 0) ? (l1 - 1) : 0;
    const float A1 = aL[pf][2 + l1];
    const float A2 = (tl > 0) ? aL[pf][2 + l2] : NEG_INF;
    ws[n] = -lae(A1, A2);
  }
}

// Fixed-order reduction -> deterministic scalar
__global__ void ctc_reduce_kernel(const float* __restrict__ ws,
                                  float* __restrict__ out) {
  if (threadIdx.x == 0 && blockIdx.x == 0) {
    float s = 0.0f;
    for (int i = 0; i < N_DIM; ++i) s += ws[i];
    out[0] = s;
  }
}

extern "C" void kernel_launch(void* const* d_in, const int* in_sizes, int n_in,
                              void* d_out, int out_size, void* d_ws, size_t ws_size,
                              hipStream_t stream) {
  (void)in_sizes; (void)n_in; (void)out_size; (void)ws_size;
  const float* log_probs      = (const float*)d_in[0];
  const int*   targets        = (const int*)d_in[1];
  const int*   input_lengths  = (const int*)d_in[2];
  const int*   target_lengths = (const int*)d_in[3];
  float* out = (float*)d_out;
  float* ws  = (float*)d_ws;

  ctc_alpha_kernel<<<N_DIM, BLOCK, 0, stream>>>(log_probs, targets,
                                                input_lengths, target_lengths, ws);
  ctc_reduce_kernel<<<1, 32, 0, stream>>>(ws, out);
}